// GraphConvWL_29300266893372
// MI455X (gfx1250) — compile-verified
//
#include <hip/hip_runtime.h>

#define D 64   // D_IN == D_OUT == 64

typedef __attribute__((ext_vector_type(2))) float v2f;
typedef __attribute__((ext_vector_type(8))) float v8f;

// ---------------- Kernel 1: zero the aggregation workspace ----------------
__global__ void GraphConv_zero_f4(float4* __restrict__ p, int n4) {
    int i = blockIdx.x * blockDim.x + threadIdx.x;
    if (i < n4) p[i] = make_float4(0.f, 0.f, 0.f, 0.f);
}

// ---------------- Kernel 2: edge gather/scale/scatter ----------------
// One wave32 per edge: lane l handles dims {2l, 2l+1}. feat row read is one
// coalesced 256B burst per wave; scatter is 64 coalesced f32 atomics into agg
// (agg is 12.8 MB -> L2 resident on MI455X's 192 MB L2).
__global__ __launch_bounds__(256) void GraphConv_edge_scatter(
    const float* __restrict__ feat, const int* __restrict__ src,
    const int* __restrict__ dst, const float* __restrict__ ew,
    float* __restrict__ agg, int n_edges) {
    int tid = blockIdx.x * blockDim.x + threadIdx.x;
    int e = tid >> 5;
    if (e >= n_edges) return;
    int d2 = (tid & 31) << 1;
    int s = src[e];
    int t = dst[e];
    float w = ew[e];
    const float2 f = *reinterpret_cast<const float2*>(feat + (size_t)s * D + d2);
    atomicAdd(agg + (size_t)t * D + d2 + 0, f.x * w);
    atomicAdd(agg + (size_t)t * D + d2 + 1, f.y * w);
}

// ---------------- Kernel 3: fused dual GEMM via V_WMMA_F32_16X16X4_F32 ----
// out = agg @ W_neigh + feat @ W_self^T + b_neigh
// One wave per 16-node tile, covering all 64 output columns (4 C tiles).
// A 16x4 f32 fragment layout (ISA 7.12.2): lanes 0-15 hold K=k,k+1 of row
// m=lane; lanes 16-31 hold K=k+2,k+3. B 4x16 mirrors with lane = column.
// C 16x16: lane -> n = lane&15, VGPR v -> m = v + 8*(lane>>4).
__global__ __launch_bounds__(256) void GraphConv_fused_gemm(
    const float* __restrict__ agg, const float* __restrict__ feat,
    const float* __restrict__ Wn, const float* __restrict__ bn,
    const float* __restrict__ Ws, float* __restrict__ out, int n_nodes) {
    const int wave = (blockIdx.x * blockDim.x + threadIdx.x) >> 5;
    const int lane = threadIdx.x & 31;
    const int row0 = wave * 16;
    if (row0 >= n_nodes) return;        // wave-uniform: EXEC stays all-ones
    const int half = lane >> 4;         // 0: K=k,k+1   1: K=k+2,k+3
    const int lr   = lane & 15;         // A row / B,C column within tile

    // C init = bias broadcast down rows
    v8f c0, c1, c2, c3;
#pragma unroll
    for (int v = 0; v < 8; ++v) {
        c0[v] = bn[ 0 + lr];
        c1[v] = bn[16 + lr];
        c2[v] = bn[32 + lr];
        c3[v] = bn[48 + lr];
    }

    const float* aggRow  = agg  + (size_t)(row0 + lr) * D;
    const float* featRow = feat + (size_t)(row0 + lr) * D;

#pragma unroll
    for (int k = 0; k < D; k += 4) {
        const int kk = k + 2 * half;

        // A fragments (float2 per lane, contiguous 8B loads)
        v2f a_agg, a_ft;
        a_agg[0] = aggRow[kk];   a_agg[1] = aggRow[kk + 1];
        a_ft[0]  = featRow[kk];  a_ft[1]  = featRow[kk + 1];

        // B fragments, neighbor GEMM: B[kq][n] = Wn[kq*D + n]
        v2f b0, b1, b2, b3;
        b0[0] = Wn[(kk    )*D +  0 + lr];  b0[1] = Wn[(kk + 1)*D +  0 + lr];
        b1[0] = Wn[(kk    )*D + 16 + lr];  b1[1] = Wn[(kk + 1)*D + 16 + lr];
        b2[0] = Wn[(kk    )*D + 32 + lr];  b2[1] = Wn[(kk + 1)*D + 32 + lr];
        b3[0] = Wn[(kk    )*D + 48 + lr];  b3[1] = Wn[(kk + 1)*D + 48 + lr];

        c0 = __builtin_amdgcn_wmma_f32_16x16x4_f32(false, a_agg, false, b0,
                                                   (short)0, c0, false, false);
        c1 = __builtin_amdgcn_wmma_f32_16x16x4_f32(false, a_agg, false, b1,
                                                   (short)0, c1, false, false);
        c2 = __builtin_amdgcn_wmma_f32_16x16x4_f32(false, a_agg, false, b2,
                                                   (short)0, c2, false, false);
        c3 = __builtin_amdgcn_wmma_f32_16x16x4_f32(false, a_agg, false, b3,
                                                   (short)0, c3, false, false);

        // B fragments, self GEMM: B[kq][n] = Ws[n*D + kq] (transposed access,
        // contiguous float2 per lane)
        v2f s0, s1, s2, s3;
        s0[0] = Ws[( 0 + lr)*D + kk];  s0[1] = Ws[( 0 + lr)*D + kk + 1];
        s1[0] = Ws[(16 + lr)*D + kk];  s1[1] = Ws[(16 + lr)*D + kk + 1];
        s2[0] = Ws[(32 + lr)*D + kk];  s2[1] = Ws[(32 + lr)*D + kk + 1];
        s3[0] = Ws[(48 + lr)*D + kk];  s3[1] = Ws[(48 + lr)*D + kk + 1];

        c0 = __builtin_amdgcn_wmma_f32_16x16x4_f32(false, a_ft, false, s0,
                                                   (short)0, c0, false, false);
        c1 = __builtin_amdgcn_wmma_f32_16x16x4_f32(false, a_ft, false, s1,
                                                   (short)0, c1, false, false);
        c2 = __builtin_amdgcn_wmma_f32_16x16x4_f32(false, a_ft, false, s2,
                                                   (short)0, c2, false, false);
        c3 = __builtin_amdgcn_wmma_f32_16x16x4_f32(false, a_ft, false, s3,
                                                   (short)0, c3, false, false);
    }

    // Store C tiles: lane writes column lr of rows m = v + 8*half
#pragma unroll
    for (int v = 0; v < 8; ++v) {
        float* o = out + (size_t)(row0 + v + 8 * half) * D;
        o[ 0 + lr] = c0[v];
        o[16 + lr] = c1[v];
        o[32 + lr] = c2[v];
        o[48 + lr] = c3[v];
    }
}

extern "C" void kernel_launch(void* const* d_in, const int* in_sizes, int n_in,
                              void* d_out, int out_size, void* d_ws, size_t ws_size,
                              hipStream_t stream) {
    const float* feat = (const float*)d_in[0];
    const int*   src  = (const int*)d_in[1];
    const int*   dst  = (const int*)d_in[2];
    const float* ew   = (const float*)d_in[3];
    const float* Wn   = (const float*)d_in[4];
    const float* bn   = (const float*)d_in[5];
    const float* Ws   = (const float*)d_in[6];
    float* out = (float*)d_out;
    float* agg = (float*)d_ws;          // [n_nodes, 64] f32 scratch

    const int n_nodes = in_sizes[0] / D;
    const int n_edges = in_sizes[1];

    // 1) zero agg
    const int n4 = (n_nodes * D) / 4;
    GraphConv_zero_f4<<<(n4 + 255) / 256, 256, 0, stream>>>((float4*)agg, n4);

    // 2) edge scatter: one wave per edge
    const long long sthreads = (long long)n_edges * 32;
    const int sblocks = (int)((sthreads + 255) / 256);
    GraphConv_edge_scatter<<<sblocks, 256, 0, stream>>>(feat, src, dst, ew,
                                                        agg, n_edges);

    // 3) fused WMMA GEMM: one wave per 16 nodes, 8 waves per block
    const int waves   = (n_nodes + 15) / 16;
    const int gblocks = (waves + 7) / 8;
    GraphConv_fused_gemm<<<gblocks, 256, 0, stream>>>(agg, feat, Wn, bn, Ws,
                                                      out, n_nodes);
}